// GatedBlock_1580547971475
// MI455X (gfx1250) — compile-verified
//
#include <hip/hip_runtime.h>
#include <stdint.h>
#include <stddef.h>

// ---------------------------------------------------------------------------
// GatedBlock: 2-tap dilated depthwise gate, fp32, memory-bound streaming.
//   x_in: (8, 256, 4096, 8)  -> per (b,c) row = 32768 contiguous floats
//   out : (8, 256, 4094, 8)  -> per (b,c) row = 32752 contiguous floats
//   out[i] = ow*( tanh(tw0*x[i]+tw1*x[i+16]+tb) * sigmoid(sw0*x[i]+sw1*x[i+16]+sb)
//                 + x[i] ) + ob
// Strategy: one workgroup per (b,c) row; double-buffered async global->LDS
// staging (CDNA5 GLOBAL_LOAD_ASYNC_TO_LDS_B128 + s_wait_asynccnt), compute
// from LDS so each input byte crosses HBM exactly once, NT b128 stores.
// Roofline: ~537 MB total traffic @ 23.3 TB/s => ~23 us. Activations use the
// gfx1250 V_TANH_F32 trans pipe (sigmoid via exact identity
// sigmoid(x)=0.5*(1+tanh(x/2))) to stay well under the VALU roofline.
// WMMA is inapplicable (per-channel diagonal weights, no matmul structure).
// ---------------------------------------------------------------------------

#define W_DIM     256
#define ROW_IN    32768              // 4096*8 floats per (b,c) row
#define ROW_OUT   32752              // 4094*8 floats per (b,c) row
#define DIL_OFF   16                 // dilation*8 floats
#define TILE      4096               // output floats per tile
#define TILE_PAD  (TILE + DIL_OFF)   // staged input floats per tile
#define NTILES    8                  // ceil(32752/4096)
#define NTHREADS  256                // 8 waves (wave32)

typedef float v4f __attribute__((ext_vector_type(4)));
typedef int   v4i __attribute__((ext_vector_type(4)));

#if defined(__has_builtin)
#if __has_builtin(__builtin_amdgcn_global_load_async_to_lds_b128)
#define HAVE_ASYNC_LDS 1
#endif
#if __has_builtin(__builtin_amdgcn_s_wait_asynccnt)
#define HAVE_WAIT_ASYNC 1
#endif
#if __has_builtin(__builtin_amdgcn_tanhf)
#define FAST_TANH(x) __builtin_amdgcn_tanhf(x)   // gfx1250 v_tanh_f32
#endif
#endif
#ifndef HAVE_ASYNC_LDS
#define HAVE_ASYNC_LDS 0
#endif
#ifndef HAVE_WAIT_ASYNC
#define HAVE_WAIT_ASYNC 0
#endif
#ifndef FAST_TANH
#define FAST_TANH(x) tanhf(x)
#endif

// Builtin signature (revealed by clang diagnostic):
//   void __builtin_amdgcn_global_load_async_to_lds_b128(
//       int4 addrspace(1)* src, int4 addrspace(3)* dst, Imm int off, Imm int cpol)
typedef __attribute__((address_space(1))) v4i gv4i;
typedef __attribute__((address_space(3))) v4i lv4i;

__device__ __forceinline__ void copy16_async(const float* g, float* l) {
#if HAVE_ASYNC_LDS
  // 16B per lane, global -> LDS, tracked by ASYNCcnt
  __builtin_amdgcn_global_load_async_to_lds_b128((gv4i*)g, (lv4i*)l, 0, 0);
#else
  *(v4f*)l = *(const v4f*)g;
#endif
}

#if HAVE_WAIT_ASYNC
#define WAIT_ASYNC(n) __builtin_amdgcn_s_wait_asynccnt(n)
#else
#define WAIT_ASYNC(n) asm volatile("s_wait_asynccnt %0" :: "n"(n) : "memory")
#endif

// Stage TILE_PAD floats (tile + 16-float dilation overlap) into LDS.
// Exactly 5 async instructions per wave regardless of lane id (uniform
// ASYNCcnt across all waves): rounds 0..3 cover TILE floats; round 4 covers
// the 16-float overlap, with lds+global addresses clamped identically so
// surplus lanes redundantly rewrite the same last float4 (same value).
__device__ __forceinline__ void issue_tile(const float* __restrict__ inRow,
                                           float* buf, int tileIn, int tid) {
#pragma unroll
  for (int r = 0; r < 5; ++r) {
    int f = (r * NTHREADS + tid) * 4;
    if (f > TILE_PAD - 4) f = TILE_PAD - 4;          // only clamps in round 4
    int gf = tileIn + f;
    if (gf > ROW_IN - 4) gf = ROW_IN - 4;            // stay inside this row
    copy16_async(inRow + gf, buf + f);
  }
}

__global__ __launch_bounds__(NTHREADS)
void gated_block_kernel(const float* __restrict__ x_in,
                        const float* __restrict__ tanh_w,
                        const float* __restrict__ tanh_b,
                        const float* __restrict__ sigmoid_w,
                        const float* __restrict__ sigmoid_b,
                        const float* __restrict__ out_w,
                        const float* __restrict__ out_b,
                        float* __restrict__ out) {
  __shared__ float buf[2][TILE_PAD];

  const int tid = threadIdx.x;
  const int row = blockIdx.x;            // row = b*256 + c
  const int c   = row & (W_DIM - 1);

  const float* inRow  = x_in + (size_t)row * ROW_IN;
  float*       outRow = out  + (size_t)row * ROW_OUT;

  // Block-uniform channel parameters -> scalar loads.
  // Sigmoid params pre-halved: sigmoid(x) = 0.5*(1 + tanh(0.5*x)).
  const float tw0 = tanh_w[2 * c + 0];
  const float tw1 = tanh_w[2 * c + 1];
  const float tbv = tanh_b[c];
  const float sw0h = 0.5f * sigmoid_w[2 * c + 0];
  const float sw1h = 0.5f * sigmoid_w[2 * c + 1];
  const float sbvh = 0.5f * sigmoid_b[c];
  const float owv = out_w[c];
  const float obv = out_b[c];

  // Prologue: prefetch tiles 0 and 1 (5 async ops each per wave).
  issue_tile(inRow, &buf[0][0], 0, tid);
  issue_tile(inRow, &buf[1][0], TILE, tid);

  for (int t = 0; t < NTILES; ++t) {
    float* bufA = &buf[t & 1][0];

    // Wait for this wave's loads of buf[t&1]; 5 newer ops (next tile) may
    // remain in flight. On the last tile nothing newer exists -> wait 0.
    if (t < NTILES - 1) { WAIT_ASYNC(5); } else { WAIT_ASYNC(0); }
    __syncthreads();   // all waves' LDS writes for this tile visible

    const int tileOut = t * TILE;
#pragma unroll
    for (int r = 0; r < TILE / (NTHREADS * 4); ++r) {
      const int j = (r * NTHREADS + tid) * 4;
      if (tileOut + j < ROW_OUT) {            // ROW_OUT % 4 == 0
        v4f a = *(const v4f*)(bufA + j);            // x0
        v4f b = *(const v4f*)(bufA + j + DIL_OFF);  // x1
        v4f o;
#pragma unroll
        for (int k = 0; k < 4; ++k) {
          float x0 = a[k], x1 = b[k];
          float pa = fmaf(tw1, x1, fmaf(tw0, x0, tbv));    // tanh pre-act
          float ps = fmaf(sw1h, x1, fmaf(sw0h, x0, sbvh)); // 0.5*sigmoid pre-act
          float sg = fmaf(0.5f, FAST_TANH(ps), 0.5f);      // sigmoid, exact identity
          float g  = FAST_TANH(pa) * sg;
          o[k] = fmaf(owv, g + x0, obv);
        }
        __builtin_nontemporal_store(o, (v4f*)(outRow + tileOut + j));
      }
    }
    __syncthreads();   // all waves done reading bufA before it is reloaded

    if (t + 2 < NTILES) issue_tile(inRow, bufA, (t + 2) * TILE, tid);
  }
}

extern "C" void kernel_launch(void* const* d_in, const int* in_sizes, int n_in,
                              void* d_out, int out_size, void* d_ws, size_t ws_size,
                              hipStream_t stream) {
  (void)in_sizes; (void)n_in; (void)out_size; (void)d_ws; (void)ws_size;
  const float* x_in      = (const float*)d_in[0];
  const float* tanh_w    = (const float*)d_in[1];
  const float* tanh_b    = (const float*)d_in[2];
  const float* sigmoid_w = (const float*)d_in[3];
  const float* sigmoid_b = (const float*)d_in[4];
  const float* out_w     = (const float*)d_in[5];
  const float* out_b     = (const float*)d_in[6];
  float* out = (float*)d_out;

  dim3 grid(8 * W_DIM);   // one workgroup per (b, c) row
  gated_block_kernel<<<grid, NTHREADS, 0, stream>>>(
      x_in, tanh_w, tanh_b, sigmoid_w, sigmoid_b, out_w, out_b, out);
}